// EstimatorQNN_65481071405621
// MI455X (gfx1250) — compile-verified
//
#include <hip/hip_runtime.h>
#include <math.h>

// ---------------------------------------------------------------------------
// EstimatorQNN (2 qubits, 2 layers) on MI455X / gfx1250.
//
// All variational gates are batch-shared -> circuit after encoding is a fixed
// 4x4 complex unitary U(weights). Encoded state is REAL:
//   psi = [c0c1, c0s1, s0c1, s0s1],  ci=cos(xi*pi/2), si=sin(xi*pi/2)
//   <Z0> = psi^T A psi,  A = Re(U^H Z U)  (4x4 real symmetric, 10 coeffs)
//
// Memory-bound: 12 B/sample * 4.19M = 50 MB -> ~2.2 us at 23.3 TB/s; the goal
// is minimal dynamic instructions per sample so waves stay available to hide
// HBM latency.
//
// WMMA mapping (V_WMMA_F32_16X16X4_F32): D = Psi[16x4] x (A Psi^T)[4x16],
// diag(D) = <Z0> for 16 samples. Both fragments are built in-register:
//   A-frag (doc layout):  lanes 0-15 hold (K0,K1)=(psi0,psi1),
//                         lanes 16-31 hold (K2,K3)=(psi2,psi3)
//   B-frag (same K split): v0 -> g0|g2, v1 -> g1|g3, g = A*psi per lane
// Diagonal extraction: D[n][n] sits in lane n (n<8, vgpr n) and lane 24+(n-8)
// (n>=8, vgpr n-8) -> 7-deep cndmask chain + direct contiguous store.
// No LDS, no shuffles, no cross-lane waits.
// ---------------------------------------------------------------------------

typedef __attribute__((ext_vector_type(2))) float v2f;
typedef __attribute__((ext_vector_type(8))) float v8f;

// ----------------------------- setup kernel --------------------------------

struct cd { double re, im; };
__device__ __forceinline__ cd cmul(cd a, cd b) {
  return { a.re * b.re - a.im * b.im, a.re * b.im + a.im * b.re };
}

__device__ void gate_rx(double t, cd g[4]) {
  double c = cos(t * 0.5), s = sin(t * 0.5);
  g[0] = { c, 0.0 }; g[1] = { 0.0, -s };
  g[2] = { 0.0, -s }; g[3] = { c, 0.0 };
}
__device__ void gate_rz(double t, cd g[4]) {
  double c = cos(t * 0.5), s = sin(t * 0.5);
  g[0] = { c, -s }; g[1] = { 0.0, 0.0 };
  g[2] = { 0.0, 0.0 }; g[3] = { c, s };
}
__device__ void gate_ry(double t, cd g[4]) {
  double c = cos(t * 0.5), s = sin(t * 0.5);
  g[0] = { c, 0.0 }; g[1] = { -s, 0.0 };
  g[2] = { s, 0.0 }; g[3] = { c, 0.0 };
}

// U <- (g (x) I2) * U    (gate on qubit 0; index i = 2*q0 + q1)
__device__ void apply_q0(cd U[16], const cd g[4]) {
  cd nU[16];
  for (int a = 0; a < 2; ++a)
    for (int b = 0; b < 2; ++b)
      for (int col = 0; col < 4; ++col) {
        cd t0 = cmul(g[a * 2 + 0], U[(0 * 2 + b) * 4 + col]);
        cd t1 = cmul(g[a * 2 + 1], U[(1 * 2 + b) * 4 + col]);
        nU[(a * 2 + b) * 4 + col] = { t0.re + t1.re, t0.im + t1.im };
      }
  for (int i = 0; i < 16; ++i) U[i] = nU[i];
}

// U <- (I2 (x) g) * U    (gate on qubit 1)
__device__ void apply_q1(cd U[16], const cd g[4]) {
  cd nU[16];
  for (int a = 0; a < 2; ++a)
    for (int b = 0; b < 2; ++b)
      for (int col = 0; col < 4; ++col) {
        cd t0 = cmul(g[b * 2 + 0], U[(a * 2 + 0) * 4 + col]);
        cd t1 = cmul(g[b * 2 + 1], U[(a * 2 + 1) * 4 + col]);
        nU[(a * 2 + b) * 4 + col] = { t0.re + t1.re, t0.im + t1.im };
      }
  for (int i = 0; i < 16; ++i) U[i] = nU[i];
}

// wsA layout: [A00,A01,A02,A03,A11,A12,A13,A22,A23,A33]
__global__ void qnn_setup_kernel(const float* __restrict__ weights,
                                 float* __restrict__ wsA) {
  if (threadIdx.x != 0) return;
  double w[12];
  for (int i = 0; i < 12; ++i) w[i] = (double)weights[i];
  cd U[16];
  for (int i = 0; i < 16; ++i) U[i] = { (i % 5 == 0) ? 1.0 : 0.0, 0.0 };
  cd g[4];
  for (int layer = 0; layer < 2; ++layer) {
    int o = 6 * layer;
    gate_rx(w[o + 0], g); apply_q0(U, g);
    gate_rz(w[o + 1], g); apply_q0(U, g);
    gate_ry(w[o + 2], g); apply_q0(U, g);
    gate_rx(w[o + 3], g); apply_q1(U, g);
    gate_rz(w[o + 4], g); apply_q1(U, g);
    gate_ry(w[o + 5], g); apply_q1(U, g);
    // CNOT (control q0, target q1): swap rows 2 and 3 of U
    for (int col = 0; col < 4; ++col) {
      cd tmp = U[2 * 4 + col];
      U[2 * 4 + col] = U[3 * 4 + col];
      U[3 * 4 + col] = tmp;
    }
  }
  // A_jk = sum_r z_r * (ReU_rj*ReU_rk + ImU_rj*ImU_rk), z = (+1,+1,-1,-1)
  const double zsg[4] = { 1.0, 1.0, -1.0, -1.0 };
  int p = 0;
  for (int j = 0; j < 4; ++j)
    for (int k = j; k < 4; ++k) {
      double acc = 0.0;
      for (int r = 0; r < 4; ++r)
        acc += zsg[r] * (U[r * 4 + j].re * U[r * 4 + k].re +
                         U[r * 4 + j].im * U[r * 4 + k].im);
      wsA[p++] = (float)acc;
    }
}

// ------------------------------ main kernel --------------------------------

__global__ void __launch_bounds__(256)
qnn_main_kernel(const float* __restrict__ in,   // [B,2]
                const float* __restrict__ wsA,  // 10 floats, sym A
                float* __restrict__ out,        // [B]
                int B, int NT) {
  const int lane = threadIdx.x & 31;
  const int wv   = (blockIdx.x * blockDim.x + threadIdx.x) >> 5;
  const int nwv  = (gridDim.x * blockDim.x) >> 5;

  // Uniform coefficients -> scalar loads / SGPRs
  const float A00 = wsA[0], A01 = wsA[1], A02 = wsA[2], A03 = wsA[3];
  const float A11 = wsA[4], A12 = wsA[5], A13 = wsA[6];
  const float A22 = wsA[7], A23 = wsA[8], A33 = wsA[9];

  const bool lo = (lane < 16);
  const bool diagLane = (lane < 8) || (lane >= 24);
  const int smax = B - 1;
  const int idx = lane & 7;

  for (int t = wv; t < NT; t += nwv) {       // uniform per wave -> EXEC all 1s
    int s = t * 16 + (lane & 15);
    int sc = (s < B) ? s : smax;             // clamp, no divergence
    float2 x = ((const float2*)in)[sc];      // global_load_b64, coalesced

    // hardware sin/cos take turns*2pi: sin(x*pi/2) = hwsin(x*0.25)
    float a0 = x.x * 0.25f;
    float a1 = x.y * 0.25f;
    float s0 = __builtin_amdgcn_sinf(a0);
    float c0 = __builtin_amdgcn_cosf(a0);
    float s1 = __builtin_amdgcn_sinf(a1);
    float c1 = __builtin_amdgcn_cosf(a1);

    float p0 = c0 * c1, p1 = c0 * s1, p2 = s0 * c1, p3 = s0 * s1;

    // g = A * psi (symmetric A)
    float g0 = A00 * p0 + A01 * p1 + A02 * p2 + A03 * p3;
    float g1 = A01 * p0 + A11 * p1 + A12 * p2 + A13 * p3;
    float g2 = A02 * p0 + A12 * p1 + A22 * p2 + A23 * p3;
    float g3 = A03 * p0 + A13 * p1 + A23 * p2 + A33 * p3;

    v2f af, bf;
    af.x = lo ? p0 : p2;  af.y = lo ? p1 : p3;   // A-frag: K{0|2}, K{1|3}
    bf.x = lo ? g0 : g2;  bf.y = lo ? g1 : g3;   // B-frag: same K split
    v8f cz = {};
    // D[b][n] = psi_b . g_n = psi_b^T A psi_n ; diag = result
    v8f d = __builtin_amdgcn_wmma_f32_16x16x4_f32(
        false, af, false, bf, (short)0, cz, false, false);

    // diag pick: vgpr (lane&7); valid in lanes 0-7 (y_0..7) and 24-31 (y_8..15)
    float y = d[0];
    y = (idx == 1) ? d[1] : y;
    y = (idx == 2) ? d[2] : y;
    y = (idx == 3) ? d[3] : y;
    y = (idx == 4) ? d[4] : y;
    y = (idx == 5) ? d[5] : y;
    y = (idx == 6) ? d[6] : y;
    y = (idx == 7) ? d[7] : y;

    if (diagLane && s < B) out[s] = y;       // contiguous 64B per wave-iter
  }
}

// ------------------------------- launcher ----------------------------------

extern "C" void kernel_launch(void* const* d_in, const int* in_sizes, int n_in,
                              void* d_out, int out_size, void* d_ws, size_t ws_size,
                              hipStream_t stream) {
  const float* inp = (const float*)d_in[0];   // [B,2] float32
  const float* wts = (const float*)d_in[1];   // [12]  float32
  float* out = (float*)d_out;                 // [B]   float32
  float* wsA = (float*)d_ws;                  // 10 floats scratch

  int B = in_sizes[0] / 2;

  qnn_setup_kernel<<<1, 32, 0, stream>>>(wts, wsA);

  int NT = (B + 15) / 16;                     // 16-sample tiles
  int wavesNeeded = (NT + 7) / 8;             // ~8 tiles per wave
  int blocks = (wavesNeeded + 7) / 8;         // 8 waves per 256-thread block
  if (blocks < 1) blocks = 1;
  qnn_main_kernel<<<blocks, 256, 0, stream>>>(inp, wsA, out, B, NT);
}